// MecFormer_80418967650857
// MI455X (gfx1250) — compile-verified
//
#include <hip/hip_runtime.h>
#include <hip/hip_bf16.h>
#include <math.h>

// Problem constants (match reference)
#define B_  64
#define S_  1024
#define F_  1024
#define D_  512
#define E_  5
#define T_  (B_ * S_)          // 65536 tokens
#define PITCH 1028             // LDS row pitch in floats (1028 % 64 == 4 -> conflict-free b64 reads)

typedef float v2f __attribute__((ext_vector_type(2)));
typedef float v8f __attribute__((ext_vector_type(8)));

static __device__ __forceinline__ v8f wmma_f32_k4(v2f a, v2f b, v8f c) {
    // V_WMMA_F32_16X16X4_F32 : D = A(16x4) * B(4x16) + C, fp32 exact
    return __builtin_amdgcn_wmma_f32_16x16x4_f32(false, a, false, b, (short)0, c, false, false);
}

// ---------------------------------------------------------------------------
// Kernel 1: fused router. One block = 16 tokens. Computes
//   H1 = relu(X @ gw0^T + gb0); H2 = relu(H1 @ gw1^T + gb1);
//   logits = H2 @ gw2^T + gb2; softmax over E; partial sum over the 16 tokens
// entirely in LDS; writes 5 partial sums per block to ws.
// ---------------------------------------------------------------------------
__global__ __launch_bounds__(256) void router_kernel(
    const float* __restrict__ X,
    const float* __restrict__ gw0, const float* __restrict__ gb0,
    const float* __restrict__ gw1, const float* __restrict__ gb1,
    const float* __restrict__ gw2, const float* __restrict__ gb2,
    float* __restrict__ part)            // [gridDim.x * 5]
{
    __shared__ float ldsA[16 * PITCH];   // X tile, later H2 tile
    __shared__ float ldsB[16 * PITCH];   // H1 tile
    __shared__ float lgt[16 * 5];        // logits -> probs

    const int tid = threadIdx.x;
    const int tt  = blockIdx.x;          // token tile index, token base = tt*16
    const long tok0 = (long)tt * 16;

    // ---- stage X tile [16][1024] into LDS (coalesced) ----
    for (int i = 0; i < 64; ++i) {
        int idx = tid + i * 256;
        int r = idx >> 10, c = idx & 1023;
        ldsA[r * PITCH + c] = X[(tok0 + r) * (long)F_ + c];
    }
    __syncthreads();

    const int w    = tid >> 5;           // wave 0..7
    const int l    = tid & 31;
    const int half = l >> 4;
    const int n    = l & 15;             // also the A-matrix row m

    // ---- layer 1: H1 = relu(X @ gw0^T + gb0) ----
    // four n-tiles per k-loop: shared A fragment, 4 independent WMMA chains
    for (int g = 0; g < 2; ++g) {
        const int h0 = (w * 8 + 4 * g) * 16;
        const float* aRow = &ldsA[n * PITCH + 2 * half];
        const float* bRow[4];
        #pragma unroll
        for (int j = 0; j < 4; ++j)
            bRow[j] = gw0 + (long)(h0 + 16 * j + n) * F_ + 2 * half;
        v8f acc[4] = {};
        #pragma unroll 2
        for (int k0 = 0; k0 < F_; k0 += 4) {
            v2f a = *(const v2f*)(aRow + k0);
            #pragma unroll
            for (int j = 0; j < 4; ++j) {
                v2f bb = *(const v2f*)(bRow[j] + k0);
                acc[j] = wmma_f32_k4(a, bb, acc[j]);
            }
        }
        #pragma unroll
        for (int j = 0; j < 4; ++j) {
            const float bias = gb0[h0 + 16 * j + n];
            #pragma unroll
            for (int v = 0; v < 8; ++v) {
                float val = acc[j][v] + bias;
                ldsB[(v + 8 * half) * PITCH + h0 + 16 * j + n] = val > 0.f ? val : 0.f;
            }
        }
    }
    __syncthreads();

    // ---- layer 2: H2 = relu(H1 @ gw1^T + gb1)  (writes over X tile) ----
    for (int g = 0; g < 2; ++g) {
        const int h0 = (w * 8 + 4 * g) * 16;
        const float* aRow = &ldsB[n * PITCH + 2 * half];
        const float* bRow[4];
        #pragma unroll
        for (int j = 0; j < 4; ++j)
            bRow[j] = gw1 + (long)(h0 + 16 * j + n) * F_ + 2 * half;
        v8f acc[4] = {};
        #pragma unroll 2
        for (int k0 = 0; k0 < F_; k0 += 4) {
            v2f a = *(const v2f*)(aRow + k0);
            #pragma unroll
            for (int j = 0; j < 4; ++j) {
                v2f bb = *(const v2f*)(bRow[j] + k0);
                acc[j] = wmma_f32_k4(a, bb, acc[j]);
            }
        }
        #pragma unroll
        for (int j = 0; j < 4; ++j) {
            const float bias = gb1[h0 + 16 * j + n];
            #pragma unroll
            for (int v = 0; v < 8; ++v) {
                float val = acc[j][v] + bias;
                ldsA[(v + 8 * half) * PITCH + h0 + 16 * j + n] = val > 0.f ? val : 0.f;
            }
        }
    }
    __syncthreads();

    // ---- layer 3: logits[16][5], wave-parallel dots + shfl reduction ----
    // 80 (token, expert) pairs; each wave handles 10; 32 lanes stride over k.
    for (int j = 0; j < 10; ++j) {
        const int pi = w * 10 + j;       // 0..79
        const int t  = pi & 15;
        const int e  = pi >> 4;          // 0..4
        const float* g2 = gw2 + (long)e * F_;
        const float* h2 = &ldsA[t * PITCH];
        float acc = 0.f;
        #pragma unroll 4
        for (int k = l; k < F_; k += 32) acc = fmaf(h2[k], g2[k], acc);
        #pragma unroll
        for (int off = 16; off > 0; off >>= 1) acc += __shfl_xor(acc, off, 32);
        if (l == 0) lgt[t * 5 + e] = acc + gb2[e];
    }
    __syncthreads();

    // ---- softmax per token ----
    if (tid < 16) {
        const int t = tid;
        float mx = lgt[t * 5];
        #pragma unroll
        for (int e = 1; e < 5; ++e) mx = fmaxf(mx, lgt[t * 5 + e]);
        float p[5], s = 0.f;
        #pragma unroll
        for (int e = 0; e < 5; ++e) { p[e] = expf(lgt[t * 5 + e] - mx); s += p[e]; }
        const float inv = 1.f / s;
        #pragma unroll
        for (int e = 0; e < 5; ++e) lgt[t * 5 + e] = p[e] * inv;
    }
    __syncthreads();

    // ---- partial sum over this tile's 16 tokens (deterministic, no atomics) ----
    if (tid < 5) {
        float s = 0.f;
        #pragma unroll
        for (int t = 0; t < 16; ++t) s += lgt[t * 5 + tid];
        part[(long)tt * 5 + tid] = s;
    }
}

// ---------------------------------------------------------------------------
// Kernel 2: reduce per-tile partials -> gate_weights [B,E] = mean over S
// ---------------------------------------------------------------------------
__global__ __launch_bounds__(320) void gate_reduce_kernel(
    const float* __restrict__ part, float* __restrict__ gate)
{
    const int tid = threadIdx.x;         // 0..319
    if (tid >= B_ * E_) return;
    const int b = tid / E_, e = tid % E_;
    float s = 0.f;
    for (int t = 0; t < 64; ++t)         // 64 tiles of 16 tokens per batch row
        s += part[((long)b * 64 + t) * 5 + e];
    gate[b * E_ + e] = s * (1.0f / (float)S_);
}

// ---------------------------------------------------------------------------
// Kernel 3: fused merged-weight GEMM.
// Block (dTile, b): builds W_eff tile [16][1024] in LDS from
// W_base + sum_e coef[b,e]*dW[e], then out = relu(X @ W_tile^T + b_eff).
// ---------------------------------------------------------------------------
__global__ __launch_bounds__(256) void moe_gemm_kernel(
    const float* __restrict__ X,
    const float* __restrict__ gate,      // [B,E] (already written to d_out tail)
    const float* __restrict__ ohe,       // [B,E]
    const float* __restrict__ W_base, const float* __restrict__ b_base,
    const float* __restrict__ dW,     const float* __restrict__ db,
    float* __restrict__ out)
{
    __shared__ float sW[16 * PITCH];
    __shared__ float sCoef[5];
    __shared__ float sBias[16];

    const int d0  = blockIdx.x * 16;     // 0..511
    const int b   = blockIdx.y;          // 0..63
    const int tid = threadIdx.x;

    if (tid < 5) sCoef[tid] = gate[b * E_ + tid] + ohe[b * E_ + tid];
    __syncthreads();
    const float c0 = sCoef[0], c1 = sCoef[1], c2 = sCoef[2], c3 = sCoef[3], c4 = sCoef[4];

    // build merged weight tile (all reads L2-resident: ~12 MB of weights total)
    for (int i = 0; i < 64; ++i) {
        int idx = tid + i * 256;
        int r = idx >> 10, f = idx & 1023;
        long wi = (long)(d0 + r) * F_ + f;
        float wv = W_base[wi];
        wv = fmaf(c0, dW[wi                 ], wv);
        wv = fmaf(c1, dW[wi + 1L * D_ * F_ ], wv);
        wv = fmaf(c2, dW[wi + 2L * D_ * F_ ], wv);
        wv = fmaf(c3, dW[wi + 3L * D_ * F_ ], wv);
        wv = fmaf(c4, dW[wi + 4L * D_ * F_ ], wv);
        sW[r * PITCH + f] = wv;
    }
    if (tid < 16) {
        const int d = d0 + tid;
        float bv = b_base[d];
        bv = fmaf(c0, db[d], bv);
        bv = fmaf(c1, db[1 * D_ + d], bv);
        bv = fmaf(c2, db[2 * D_ + d], bv);
        bv = fmaf(c3, db[3 * D_ + d], bv);
        bv = fmaf(c4, db[4 * D_ + d], bv);
        sBias[tid] = bv;
    }
    __syncthreads();

    const int w    = tid >> 5;
    const int l    = tid & 31;
    const int half = l >> 4;
    const int n    = l & 15;
    const float bias = sBias[n];

    // each wave covers 8 of 64 token tiles; four tiles per k-loop share the
    // B fragment (LDS) and give four independent WMMA accumulation chains.
    for (int g = 0; g < 2; ++g) {
        int t0[4];
        const float* aRow[4];
        #pragma unroll
        for (int j = 0; j < 4; ++j) {
            t0[j] = (w + (4 * g + j) * 8) * 16;
            aRow[j] = X + ((long)b * S_ + t0[j] + n) * F_ + 2 * half;
        }
        const float* bRow = &sW[n * PITCH + 2 * half];
        v8f acc[4] = {};
        #pragma unroll 2
        for (int k0 = 0; k0 < F_; k0 += 4) {
            v2f bb = *(const v2f*)(bRow + k0);
            #pragma unroll
            for (int j = 0; j < 4; ++j) {
                v2f a = *(const v2f*)(aRow[j] + k0);
                acc[j] = wmma_f32_k4(a, bb, acc[j]);
            }
        }
        #pragma unroll
        for (int j = 0; j < 4; ++j) {
            #pragma unroll
            for (int v = 0; v < 8; ++v) {
                float val = acc[j][v] + bias;
                val = val > 0.f ? val : 0.f;
                out[((long)b * S_ + t0[j] + v + 8 * half) * (long)D_ + d0 + n] = val;
            }
        }
    }
}

// ---------------------------------------------------------------------------
extern "C" void kernel_launch(void* const* d_in, const int* in_sizes, int n_in,
                              void* d_out, int out_size, void* d_ws, size_t ws_size,
                              hipStream_t stream)
{
    const float* X      = (const float*)d_in[0];   // [B,S,F]
    const float* ohe    = (const float*)d_in[1];   // [B,E]
    const float* gw0    = (const float*)d_in[2];
    const float* gb0    = (const float*)d_in[3];
    const float* gw1    = (const float*)d_in[4];
    const float* gb1    = (const float*)d_in[5];
    const float* gw2    = (const float*)d_in[6];
    const float* gb2    = (const float*)d_in[7];
    const float* W_base = (const float*)d_in[8];
    const float* b_base = (const float*)d_in[9];
    const float* dW     = (const float*)d_in[10];
    const float* db     = (const float*)d_in[11];

    float* out  = (float*)d_out;                       // [B,S,D]
    float* gate = out + (long)B_ * S_ * D_;            // [B,E] tail of d_out
    float* part = (float*)d_ws;                        // [T_/16 * 5] partial sums

    const int nTiles = T_ / 16;                        // 4096

    router_kernel<<<nTiles, 256, 0, stream>>>(X, gw0, gb0, gw1, gb1, gw2, gb2, part);
    gate_reduce_kernel<<<1, 320, 0, stream>>>(part, gate);
    moe_gemm_kernel<<<dim3(D_ / 16, B_), 256, 0, stream>>>(
        X, gate, ohe, W_base, b_base, dW, db, out);
}